// GCN_34308198760738
// MI455X (gfx1250) — compile-verified
//
#include <hip/hip_runtime.h>
#include <hip/hip_bf16.h>

#define N_NODES 100000
#define N_EDGES 800000
#define DIM     128
#define DD      (DIM * DIM)
#define NCONV   3
#define LDSP    132   // padded LDS row stride (floats): conflict-free A-fragment reads

typedef float v2f __attribute__((ext_vector_type(2)));
typedef float v8f __attribute__((ext_vector_type(8)));

// ---------------------------------------------------------------------------
// GEMM: H[M x 128] = X[M x 128] @ W[128 x 128] (+ optional bias, relu)
// One block = 256 threads = 8 waves. Block owns a 16-row strip of X (in LDS),
// wave w owns output cols [16w, 16w+16). 32 x v_wmma_f32_16x16x4_f32 per tile.
// Fragment layouts per CDNA5 ISA 7.12.2.
// ---------------------------------------------------------------------------
template <bool BIAS_RELU>
__global__ __launch_bounds__(256) void gemm_wmma_f32(
    const float* __restrict__ X, const float* __restrict__ W,
    const float* __restrict__ bias, float* __restrict__ H) {
  __shared__ float xs[16 * LDSP];
  const int tid = threadIdx.x;
  const int rowBase = blockIdx.x * 16;

  // Stage 16x128 strip of X into LDS (coalesced float4 loads).
  for (int i = tid; i < 16 * 32; i += 256) {
    const int r = i >> 5, c4 = (i & 31) << 2;
    const float4 v = *(const float4*)(X + (size_t)(rowBase + r) * DIM + c4);
    float* p = &xs[r * LDSP + c4];
    p[0] = v.x; p[1] = v.y; p[2] = v.z; p[3] = v.w;
  }
  __syncthreads();

  const int wave = tid >> 5;
  const int lane = tid & 31;
  const int half = lane >> 4;   // 0: lanes 0-15, 1: lanes 16-31
  const int l15  = lane & 15;
  const int colBase = wave * 16;

  v8f c = {};
  for (int k = 0; k < DIM; k += 4) {
    // A 16x4: VGPR0 = {K=k | K=k+2}, VGPR1 = {K=k+1 | K=k+3}; M = l15
    v2f a, b;
    const float* ap = &xs[l15 * LDSP + k + 2 * half];
    a.x = ap[0];
    a.y = ap[1];
    // B 4x16: lanes 0-15 hold rows k,k+1; lanes 16-31 hold rows k+2,k+3; N = l15
    const float* bp = W + (size_t)(k + 2 * half) * DIM + colBase + l15;
    b.x = bp[0];
    b.y = bp[DIM];
    c = __builtin_amdgcn_wmma_f32_16x16x4_f32(false, a, false, b, (short)0, c,
                                              false, false);
  }

  // D 16x16: VGPR r -> row (r + 8*half), col = l15
  const int col = colBase + l15;
  for (int r = 0; r < 8; ++r) {
    float v = c[r];
    if (BIAS_RELU) v = fmaxf(v + bias[col], 0.0f);
    H[(size_t)(rowBase + r + 8 * half) * DIM + col] = v;
  }
}

// ---------------------------------------------------------------------------
// Graph plumbing kernels
// ---------------------------------------------------------------------------
__global__ void zero_kernel(float* __restrict__ p, int n) {
  int i = blockIdx.x * blockDim.x + threadIdx.x;
  if (i < n) p[i] = 0.0f;
}

__global__ void deg_kernel(const int* __restrict__ dst, float* __restrict__ deg,
                           int E) {
  int e = blockIdx.x * blockDim.x + threadIdx.x;
  if (e < E) atomicAdd(&deg[dst[e]], 1.0f);
}

// in-place: deg -> rsqrt(deg + 1)   (+1 = self loop)
__global__ void dinv_kernel(float* __restrict__ deg, int n) {
  int i = blockIdx.x * blockDim.x + threadIdx.x;
  if (i < n) deg[i] = rsqrtf(deg[i] + 1.0f);
}

// out[n][d] = conv_bias[d] + h[n][d] * dinv[n]^2   (bias + self-loop term)
__global__ void init_out_kernel(const float* __restrict__ H,
                                const float* __restrict__ dinv,
                                const float* __restrict__ bias,
                                float* __restrict__ OUT) {
  int i = blockIdx.x * blockDim.x + threadIdx.x;  // over N*D
  int n = i >> 7, d = i & 127;
  float di = dinv[n];
  OUT[i] = bias[d] + H[i] * di * di;
}

// One wave per edge: OUT[dst] += H[src] * (dinv[src]*dinv[dst])
__global__ void edge_agg_kernel(const float* __restrict__ H,
                                const int* __restrict__ src,
                                const int* __restrict__ dst,
                                const float* __restrict__ dinv,
                                float* __restrict__ OUT, int E) {
  int e = (blockIdx.x * blockDim.x + threadIdx.x) >> 5;
  int lane = threadIdx.x & 31;
  if (e >= E) return;
  int s = src[e], d = dst[e];
  float nrm = dinv[s] * dinv[d];
  const float4 v = *(const float4*)(H + (size_t)s * DIM + lane * 4);
  float* o = OUT + (size_t)d * DIM + lane * 4;
  atomicAdd(o + 0, v.x * nrm);
  atomicAdd(o + 1, v.y * nrm);
  atomicAdd(o + 2, v.z * nrm);
  atomicAdd(o + 3, v.w * nrm);
}

// Column sums / sums-of-squares (128 threads = one column each, strided rows)
__global__ void stats_kernel(const float* __restrict__ X,
                             float* __restrict__ colsum,
                             float* __restrict__ colsumsq, int N) {
  int c = threadIdx.x;
  float s = 0.0f, ss = 0.0f;
  for (int r = blockIdx.x; r < N; r += gridDim.x) {
    float v = X[(size_t)r * DIM + c];
    s += v;
    ss += v * v;
  }
  atomicAdd(&colsum[c], s);
  atomicAdd(&colsumsq[c], ss);
}

// a = ms*mean ; istd = rsqrt(E[x^2] - 2a*mean + a^2 + eps)
__global__ void finalize_stats_kernel(const float* __restrict__ colsum,
                                      const float* __restrict__ colsumsq,
                                      const float* __restrict__ ms,
                                      float* __restrict__ avec,
                                      float* __restrict__ istd, int N) {
  int c = threadIdx.x;
  float inv_n = 1.0f / (float)N;
  float mean = colsum[c] * inv_n;
  float a = ms[c] * mean;
  float var = colsumsq[c] * inv_n - 2.0f * a * mean + a * a;
  avec[c] = a;
  istd[c] = rsqrtf(var + 1e-5f);
}

// x = relu(w * (x - a) * istd + b), in place
__global__ void apply_norm_relu_kernel(float* __restrict__ X,
                                       const float* __restrict__ avec,
                                       const float* __restrict__ istd,
                                       const float* __restrict__ w,
                                       const float* __restrict__ b) {
  int i = blockIdx.x * blockDim.x + threadIdx.x;  // over N*D
  int c = i & 127;
  float v = (X[i] - avec[c]) * istd[c] * w[c] + b[c];
  X[i] = fmaxf(v, 0.0f);
}

// Head: out[n][0..1] = Y[n] @ lin_w1 + b1 ; one wave per node, shfl reduce
__global__ void head_kernel(const float* __restrict__ Y,
                            const float* __restrict__ w1,
                            const float* __restrict__ b1,
                            float* __restrict__ out, int N) {
  int n = (blockIdx.x * blockDim.x + threadIdx.x) >> 5;
  int lane = threadIdx.x & 31;
  if (n >= N) return;
  const float4 v = *(const float4*)(Y + (size_t)n * DIM + lane * 4);
  int k = lane * 4;
  float a0 = v.x * w1[(k + 0) * 2 + 0] + v.y * w1[(k + 1) * 2 + 0] +
             v.z * w1[(k + 2) * 2 + 0] + v.w * w1[(k + 3) * 2 + 0];
  float a1 = v.x * w1[(k + 0) * 2 + 1] + v.y * w1[(k + 1) * 2 + 1] +
             v.z * w1[(k + 2) * 2 + 1] + v.w * w1[(k + 3) * 2 + 1];
  for (int off = 16; off > 0; off >>= 1) {
    a0 += __shfl_xor(a0, off, 32);
    a1 += __shfl_xor(a1, off, 32);
  }
  if (lane == 0) {
    out[(size_t)n * 2 + 0] = a0 + b1[0];
    out[(size_t)n * 2 + 1] = a1 + b1[1];
  }
}

// ---------------------------------------------------------------------------
extern "C" void kernel_launch(void* const* d_in, const int* in_sizes, int n_in,
                              void* d_out, int out_size, void* d_ws,
                              size_t ws_size, hipStream_t stream) {
  const float* x       = (const float*)d_in[0];
  const int*   ei      = (const int*)d_in[1];
  const float* conv_w  = (const float*)d_in[2];
  const float* conv_b  = (const float*)d_in[3];
  const float* gn_w    = (const float*)d_in[4];
  const float* gn_b    = (const float*)d_in[5];
  const float* gn_ms   = (const float*)d_in[6];
  const float* lin_w0  = (const float*)d_in[7];
  const float* lin_b0  = (const float*)d_in[8];
  const float* lin_w1  = (const float*)d_in[9];
  const float* lin_b1  = (const float*)d_in[10];
  float*       out     = (float*)d_out;

  const int* srcE = ei;            // edge_index[0]
  const int* dstE = ei + N_EDGES;  // edge_index[1]

  const size_t ND = (size_t)N_NODES * DIM;
  float* ws       = (float*)d_ws;
  float* bufA     = ws;            // GEMM output h
  float* bufB     = ws + ND;       // aggregation output / running features
  float* dinv     = ws + 2 * ND;   // N (deg accumulator, then rsqrt)
  float* colsum   = dinv + N_NODES;
  float* colsumsq = colsum + DIM;
  float* avec     = colsumsq + DIM;
  float* istd     = avec + DIM;

  const int T = 256;
  const int gridN  = (N_NODES + T - 1) / T;
  const int gridE  = (N_EDGES + T - 1) / T;
  const int gridND = (int)(ND / T);            // 50000 (exact)
  const int gridGemm = N_NODES / 16;           // 6250 (exact)
  const int gridAgg  = (N_EDGES * 32 + T - 1) / T;  // 100000
  const int gridHead = (N_NODES * 32 + T - 1) / T;  // 12500

  // Degrees (with self loop) -> dinv = deg^{-1/2}
  zero_kernel<<<gridN, T, 0, stream>>>(dinv, N_NODES);
  deg_kernel<<<gridE, T, 0, stream>>>(dstE, dinv, N_EDGES);
  dinv_kernel<<<gridN, T, 0, stream>>>(dinv, N_NODES);

  const float* cur = x;
  for (int l = 0; l < NCONV; ++l) {
    // h = cur @ W_l   (fp32 WMMA)
    gemm_wmma_f32<false><<<gridGemm, T, 0, stream>>>(
        cur, conv_w + (size_t)l * DD, nullptr, bufA);
    // out = bias + self-loop term, then scatter-add edges
    init_out_kernel<<<gridND, T, 0, stream>>>(bufA, dinv,
                                              conv_b + (size_t)l * DIM, bufB);
    edge_agg_kernel<<<gridAgg, T, 0, stream>>>(bufA, srcE, dstE, dinv, bufB,
                                               N_EDGES);
    // GraphNorm + ReLU (single-pass column stats)
    zero_kernel<<<1, 2 * DIM, 0, stream>>>(colsum, 2 * DIM);
    stats_kernel<<<512, DIM, 0, stream>>>(bufB, colsum, colsumsq, N_NODES);
    finalize_stats_kernel<<<1, DIM, 0, stream>>>(colsum, colsumsq,
                                                 gn_ms + (size_t)l * DIM, avec,
                                                 istd, N_NODES);
    apply_norm_relu_kernel<<<gridND, T, 0, stream>>>(
        bufB, avec, istd, gn_w + (size_t)l * DIM, gn_b + (size_t)l * DIM);
    cur = bufB;
  }

  // y = relu(x @ lin_w0 + b0)  (WMMA with fused bias+relu epilogue)
  gemm_wmma_f32<true><<<gridGemm, T, 0, stream>>>(cur, lin_w0, lin_b0, bufA);
  // out = y @ lin_w1 + b1
  head_kernel<<<gridHead, T, 0, stream>>>(bufA, lin_w1, lin_b1, out, N_NODES);
}